// SparseDeConvBlock_76673756168768
// MI455X (gfx1250) — compile-verified
//
#include <hip/hip_runtime.h>
#include <hip/hip_bf16.h>

#define N_IN   200000
#define N_OUT  400000
#define KVOL   8
#define C_IN   128
#define C_OUT  96
#define BN_EPS 1e-5f

typedef float v2f __attribute__((ext_vector_type(2)));
typedef float v8f __attribute__((ext_vector_type(8)));

// ---------------------------------------------------------------------------
// Kernel 1: zero the scatter-add accumulator (d_out) and the BN stat slots.
// ---------------------------------------------------------------------------
__global__ __launch_bounds__(256) void sdc_zero_kernel(float4* __restrict__ out4,
                                                       int n4,
                                                       float* __restrict__ ws) {
    if (blockIdx.x == 0 && threadIdx.x < 2 * C_OUT) ws[threadIdx.x] = 0.0f;
    const float4 z = make_float4(0.f, 0.f, 0.f, 0.f);
    for (int i = blockIdx.x * blockDim.x + threadIdx.x; i < n4;
         i += gridDim.x * blockDim.x)
        out4[i] = z;
}

// ---------------------------------------------------------------------------
// Kernel 2: rulebook gather -> fp32 WMMA GEMM -> atomic scatter-add.
// grid = (N_IN/64, KVOL), block = 128 threads = 4 waves, each wave = 16 rows.
//
// W[k] (128x96 fp32, 48KB) is staged in LDS pair-interleaved:
//   lds[((kk>>1)*96 + n)*2 + (kk&1)] = W[k][kk][n]
// so a B fragment for WMMA step s, half-wave g, column n is the contiguous
// 8-byte pair {W[4s+2g][n], W[4s+2g+1][n]}  -> one ds_load_b64 per fragment.
//
// A fragment lane layout (ISA 16x4 f32 A): lane l<16 holds {K=4s,4s+1} of row
// M=l; lane 16+l holds {K=4s+2,4s+3} of row M=l  -> one global_load_b64 from
// the gathered feats row per K-step.
// ---------------------------------------------------------------------------
__global__ __launch_bounds__(128) void sdc_gemm_scatter_kernel(
    const float* __restrict__ feats,
    const float* __restrict__ W,
    const int*   __restrict__ in_idx,
    const int*   __restrict__ out_idx,
    float*       __restrict__ out) {
    __shared__ float lw[(C_IN / 2) * C_OUT * 2];  // 48 KB

    const int k   = blockIdx.y;
    const int tid = threadIdx.x;

    // Cooperative stage of W[k] into LDS (pair-interleaved layout).
    const float* Wg = W + (size_t)k * C_IN * C_OUT;
    for (int e = tid; e < C_IN * C_OUT; e += 128) {
        const int kk = e / C_OUT;
        const int n  = e - kk * C_OUT;
        lw[((kk >> 1) * C_OUT + n) * 2 + (kk & 1)] = Wg[e];
    }
    __syncthreads();

    const int wave = tid >> 5;
    const int lane = tid & 31;
    const int g    = lane >> 4;   // half-wave group (0 or 1)
    const int l    = lane & 15;
    const int m0   = (blockIdx.x * 4 + wave) * 16;

    // Gather: base pointer of this lane's input-feature row.
    const int in_row = in_idx[k * N_IN + m0 + l];
    const float* ap  = feats + (size_t)in_row * C_IN + 2 * g;

    v8f acc[6] = {};  // 6 x (16x16 f32) accumulators covering C_OUT = 96

    #pragma unroll 8
    for (int s = 0; s < C_IN / 4; ++s) {
        const v2f a = *(const v2f*)(ap + 4 * s);              // global_load_b64
        const float* bp = &lw[((2 * s + g) * C_OUT + l) * 2]; // LDS base
        #pragma unroll
        for (int n = 0; n < 6; ++n) {
            const v2f b = *(const v2f*)(bp + 32 * n);         // ds_load_b64
            acc[n] = __builtin_amdgcn_wmma_f32_16x16x4_f32(
                false, a, false, b, (short)0, acc[n], false, false);
        }
    }

    // Scatter-add. C/D layout: VGPR r -> M = 8*g + r, column = 16*n + l.
    int orow[8];
    #pragma unroll
    for (int r = 0; r < 8; ++r)
        orow[r] = out_idx[k * N_IN + m0 + 8 * g + r];

    #pragma unroll
    for (int n = 0; n < 6; ++n) {
        #pragma unroll
        for (int r = 0; r < 8; ++r) {
            atomicAdd(&out[(size_t)orow[r] * C_OUT + 16 * n + l], acc[n][r]);
        }
    }
}

// ---------------------------------------------------------------------------
// Kernel 3: per-channel sum / sum-of-squares partials (fp32 atomics into ws).
// blockDim = 96 (one thread per channel, coalesced row reads).
// ---------------------------------------------------------------------------
__global__ __launch_bounds__(96) void sdc_colstats_kernel(
    const float* __restrict__ out, float* __restrict__ ws, int rows_per_block) {
    const int c = threadIdx.x;
    const size_t r0 = (size_t)blockIdx.x * rows_per_block;
    float s = 0.f, q = 0.f;
    for (int i = 0; i < rows_per_block; ++i) {
        const float v = out[(r0 + i) * C_OUT + c];
        s += v;
        q += v * v;
    }
    atomicAdd(&ws[c], s);
    atomicAdd(&ws[C_OUT + c], q);
}

// ---------------------------------------------------------------------------
// Kernel 4: fold batch stats + gamma/beta into per-channel scale & shift.
// ---------------------------------------------------------------------------
__global__ void sdc_bnparam_kernel(const float* __restrict__ gamma,
                                   const float* __restrict__ beta,
                                   float* __restrict__ ws) {
    const int c = threadIdx.x;
    if (c < C_OUT) {
        const float inv_n = 1.0f / (float)N_OUT;
        const float mean  = ws[c] * inv_n;
        const float var   = ws[C_OUT + c] * inv_n - mean * mean;
        const float sc    = gamma[c] * rsqrtf(var + BN_EPS);
        ws[2 * C_OUT + c] = sc;
        ws[3 * C_OUT + c] = beta[c] - mean * sc;
    }
}

// ---------------------------------------------------------------------------
// Kernel 5: in-place normalize + ReLU, float4 vectorized.
// ---------------------------------------------------------------------------
__global__ __launch_bounds__(256) void sdc_bnapply_kernel(
    float* __restrict__ out, const float* __restrict__ ws) {
    __shared__ float sc[C_OUT], sh[C_OUT];
    if (threadIdx.x < C_OUT) {
        sc[threadIdx.x] = ws[2 * C_OUT + threadIdx.x];
        sh[threadIdx.x] = ws[3 * C_OUT + threadIdx.x];
    }
    __syncthreads();

    float4* out4 = (float4*)out;
    const int total4 = N_OUT * C_OUT / 4;  // 9.6M float4, row = 24 float4
    for (int i = blockIdx.x * blockDim.x + threadIdx.x; i < total4;
         i += gridDim.x * blockDim.x) {
        float4 v = out4[i];
        const int j = (i % 24) * 4;  // channel base of this float4
        v.x = fmaxf(v.x * sc[j + 0] + sh[j + 0], 0.f);
        v.y = fmaxf(v.y * sc[j + 1] + sh[j + 1], 0.f);
        v.z = fmaxf(v.z * sc[j + 2] + sh[j + 2], 0.f);
        v.w = fmaxf(v.w * sc[j + 3] + sh[j + 3], 0.f);
        out4[i] = v;
    }
}

// ---------------------------------------------------------------------------
extern "C" void kernel_launch(void* const* d_in, const int* in_sizes, int n_in,
                              void* d_out, int out_size, void* d_ws,
                              size_t ws_size, hipStream_t stream) {
    const float* feats   = (const float*)d_in[0];  // [N_IN, C_IN]
    const float* W       = (const float*)d_in[1];  // [K, C_IN, C_OUT]
    const float* gamma   = (const float*)d_in[2];  // [C_OUT]
    const float* beta    = (const float*)d_in[3];  // [C_OUT]
    const int*   in_idx  = (const int*)d_in[4];    // [K, N_IN]
    const int*   out_idx = (const int*)d_in[5];    // [K, N_IN]
    float*       out     = (float*)d_out;          // [N_OUT, C_OUT]
    float*       ws      = (float*)d_ws;           // >= 4*C_OUT floats used

    // 1) zero accumulator + stat slots
    sdc_zero_kernel<<<2048, 256, 0, stream>>>((float4*)out, N_OUT * C_OUT / 4,
                                              ws);

    // 2) gather -> fp32 WMMA GEMM -> atomic scatter
    dim3 grid(N_IN / 64, KVOL);  // 200000/64 = 3125 exact
    sdc_gemm_scatter_kernel<<<grid, 128, 0, stream>>>(feats, W, in_idx, out_idx,
                                                      out);

    // 3) per-channel batch statistics (800 blocks x 500 rows = 400000)
    sdc_colstats_kernel<<<800, 96, 0, stream>>>(out, ws, N_OUT / 800);

    // 4) fold into scale/shift
    sdc_bnparam_kernel<<<1, 96, 0, stream>>>(gamma, beta, ws);

    // 5) normalize + ReLU in place
    sdc_bnapply_kernel<<<2048, 256, 0, stream>>>(out, ws);
}